// Boundaries_Loss_57741540327996
// MI455X (gfx1250) — compile-verified
//
#include <hip/hip_runtime.h>
#include <hip/hip_bf16.h>

typedef __attribute__((ext_vector_type(2))) float v2f;
typedef __attribute__((ext_vector_type(8))) float v8f;

#define BT 8
#define NV 10000
#define NB 16384
#define NS 4096
#define VT (NV / 16)      // 625 vertex tiles (exact)
#define ST (NS / 16)      // 256 sample tiles per batch
#define NTILE (BT * ST)   // 2048 total tiles
#define FLT_BIG 3.402823466e38f

// Async double-buffer staging: 5 chunks x 125 vertex tiles (= 2000 float4 =
// 32000 B per chunk).  62 x B128 ops + B64 tail = 63 transfers, padded to 64
// (wave 7 duplicates the benign tail) so every wave issues exactly 8 async
// ops per chunk -> constant s_wait_asynccnt immediates.
#define CHUNK_T 125
#define NCHUNK 5
#define CHUNK_E (CHUNK_T * 16)   // 2000 float4
#define USE_ASYNC_LDS 1

// ---------------------------------------------------------------------------
// Kernel 1: Vp[b*NV + v] = (-2x, -2y, -2z, x^2+y^2+z^2)  (B-matrix operand)
// ---------------------------------------------------------------------------
__global__ __launch_bounds__(256) void prep_verts(const float* __restrict__ verts,
                                                  float4* __restrict__ Vp) {
  int i = blockIdx.x * blockDim.x + threadIdx.x;  // flat b*NV + v
  if (i >= BT * NV) return;
  float x = verts[3 * i + 0];
  float y = verts[3 * i + 1];
  float z = verts[3 * i + 2];
  Vp[i] = make_float4(-2.0f * x, -2.0f * y, -2.0f * z, x * x + y * y + z * z);
}

// ---------------------------------------------------------------------------
// Kernel 2: one wave32 per 16-sample tile, 8 waves (one batch) per block.
// Vertex operand tiles are staged global->LDS with async-LDS copies (ASYNCcnt,
// double buffered); D = A x B via V_WMMA_F32_16X16X4_F32; running element-min;
// cross-lane min; combine with ||b||^2 and mask; one partial per tile.
// ---------------------------------------------------------------------------
__global__ __launch_bounds__(256) void loss_main(const float4* __restrict__ Vp,
                                                 const float* __restrict__ bds,
                                                 const long long* __restrict__ indices,
                                                 float* __restrict__ partials) {
  __shared__ float rowmin[8][16];
#if USE_ASYNC_LDS
  __shared__ float4 buf[2][CHUNK_E];   // 2 x 32000 B
#endif

  const int wave = threadIdx.x >> 5;
  const int lane = threadIdx.x & 31;
  const int tile = blockIdx.x * 8 + wave;   // 0..2047 (grid sized exactly)
  const int b = tile / ST;                  // identical for all 8 waves
  const int stile = tile - b * ST;
  const int m = lane & 15;                  // row (A) / col (B) index in tile
  const int hi = lane >> 4;                 // half selects K={0,1} vs K={2,3}

  // Gather boundary point for sample row m (replicated across halves).
  const int s = stile * 16 + m;
  const long long vi = indices[s];
  const float* bp = bds + ((long long)b * NB + vi) * 4;
  const float bx = bp[0], by = bp[1], bz = bp[2], msk = bp[3];
  const float sqb = bx * bx + by * by + bz * bz;

  // A-matrix 16x4 f32: VGPR0 = {K0 | K2}, VGPR1 = {K1 | K3} across lane halves.
  v2f A;
  A.x = hi ? bz : bx;     // K0 = bx, K2 = bz
  A.y = hi ? 1.0f : by;   // K1 = by, K3 = 1

  v8f mn;
#pragma unroll
  for (int r = 0; r < 8; ++r) mn[r] = FLT_BIG;

  const float4* vb = Vp + b * NV;

#if USE_ASYNC_LDS
  // ---- async staging of one chunk (wave-uniform control, EXEC stays full) --
  auto stage = [&](int c) {
    const float4* src = vb + c * CHUNK_E;
    float4* dst = &buf[c & 1][0];
#pragma unroll
    for (int k = 0; k < 8; ++k) {
      const int i = wave + 8 * k;  // 0..63, wave-uniform
      if (i < 62) {
        const int e = i * 32 + lane;                       // one float4 / lane
        unsigned ldsoff = (unsigned)(size_t)(dst + e);     // AS3 offset bits
        unsigned long long ga = (unsigned long long)(size_t)(src + e);
        asm volatile("global_load_async_to_lds_b128 %0, %1, off"
                     :: "v"(ldsoff), "v"(ga) : "memory");
      } else {  // i == 62 (wave 6) and i == 63 (wave 7, benign duplicate)
        const unsigned byteoff = 62u * 32u * 16u + (unsigned)lane * 8u;
        unsigned ldsoff = (unsigned)(size_t)((char*)dst + byteoff);
        unsigned long long ga =
            (unsigned long long)(size_t)((const char*)src + byteoff);
        asm volatile("global_load_async_to_lds_b64 %0, %1, off"
                     :: "v"(ldsoff), "v"(ga) : "memory");
      }
    }
  };

  stage(0);
  for (int c = 0; c < NCHUNK; ++c) {
    if (c + 1 < NCHUNK) {
      stage(c + 1);                                   // overlap next transfer
      asm volatile("s_wait_asynccnt 8" ::: "memory"); // chunk c complete
    } else {
      asm volatile("s_wait_asynccnt 0" ::: "memory");
    }
    __syncthreads();  // publish every wave's staged LDS writes

    const float4* bufc = &buf[c & 1][0];
#pragma unroll 5
    for (int tt = 0; tt < CHUNK_T; ++tt) {
      float4 vp = bufc[tt * 16 + m];                  // ds_load_b128
      v2f B;
      B.x = hi ? vp.z : vp.x;                         // -2vx / -2vz
      B.y = hi ? vp.w : vp.y;                         // -2vy / ||v||^2
      v8f cz = {};
      v8f d = __builtin_amdgcn_wmma_f32_16x16x4_f32(
          false, A, false, B, (short)0, cz, false, false);
#pragma unroll
      for (int r = 0; r < 8; ++r) mn[r] = fminf(mn[r], d[r]);
    }
    __syncthreads();  // chunk fully consumed before its buffer is restaged
  }
#else
  // Fallback: direct per-wave global loads.
#pragma unroll 5
  for (int t = 0; t < VT; ++t) {
    float4 vp = vb[t * 16 + m];
    __builtin_prefetch(&vb[(t + 8) * 16 + m], 0, 3);
    v2f B;
    B.x = hi ? vp.z : vp.x;
    B.y = hi ? vp.w : vp.y;
    v8f cz = {};
    v8f d = __builtin_amdgcn_wmma_f32_16x16x4_f32(
        false, A, false, B, (short)0, cz, false, false);
#pragma unroll
    for (int r = 0; r < 8; ++r) mn[r] = fminf(mn[r], d[r]);
  }
#endif

  // Min over the 16 columns held by each 16-lane half (masks 1,2,4,8 stay
  // inside a half on wave32).
#pragma unroll
  for (int r = 0; r < 8; ++r) {
    float v = mn[r];
    v = fminf(v, __shfl_xor(v, 1, 32));
    v = fminf(v, __shfl_xor(v, 2, 32));
    v = fminf(v, __shfl_xor(v, 4, 32));
    v = fminf(v, __shfl_xor(v, 8, 32));
    mn[r] = v;
  }
  if (m == 0) {  // lanes 0 and 16 publish their half's 8 row-mins
#pragma unroll
    for (int r = 0; r < 8; ++r) rowmin[wave][hi * 8 + r] = mn[r];
  }
  __syncthreads();

  // dist_min[s] = rowmin + ||b||^2 ; masked contribution; sum 16 samples.
  float val = (hi == 0) ? (rowmin[wave][m] + sqb) * msk : 0.0f;
  val += __shfl_xor(val, 1, 32);
  val += __shfl_xor(val, 2, 32);
  val += __shfl_xor(val, 4, 32);
  val += __shfl_xor(val, 8, 32);
  val += __shfl_xor(val, 16, 32);
  if (lane == 0) partials[tile] = val;
}

// ---------------------------------------------------------------------------
// Kernel 3: deterministic final reduction of 2048 partials -> scalar mean.
// ---------------------------------------------------------------------------
__global__ __launch_bounds__(256) void reduce_final(const float* __restrict__ partials,
                                                    float* __restrict__ out,
                                                    int n, float scale) {
  __shared__ float sm[256];
  float s = 0.0f;
  for (int i = threadIdx.x; i < n; i += 256) s += partials[i];
  sm[threadIdx.x] = s;
  __syncthreads();
  for (int st = 128; st > 0; st >>= 1) {
    if (threadIdx.x < st) sm[threadIdx.x] += sm[threadIdx.x + st];
    __syncthreads();
  }
  if (threadIdx.x == 0) out[0] = sm[0] * scale;
}

// ---------------------------------------------------------------------------
extern "C" void kernel_launch(void* const* d_in, const int* in_sizes, int n_in,
                              void* d_out, int out_size, void* d_ws, size_t ws_size,
                              hipStream_t stream) {
  const float* verts = (const float*)d_in[0];           // (8,10000,3) f32
  const float* bds = (const float*)d_in[1];             // (8,16384,4) f32
  // d_in[2] = pix_to_face (unused by the reference math)
  const long long* indices = (const long long*)d_in[3]; // (4096,) int64

  float4* Vp = (float4*)d_ws;                                      // 1.28 MB
  float* partials = (float*)((char*)d_ws + (size_t)BT * NV * sizeof(float4));

  prep_verts<<<(BT * NV + 255) / 256, 256, 0, stream>>>(verts, Vp);
  loss_main<<<NTILE / 8, 256, 0, stream>>>(Vp, bds, indices, partials);
  reduce_final<<<1, 256, 0, stream>>>(partials, (float*)d_out, NTILE,
                                      1.0f / (float)(BT * NS));
}